// TopK_27419071218495
// MI455X (gfx1250) — compile-verified
//
#include <hip/hip_runtime.h>
#include <hip/hip_bf16.h>

typedef __attribute__((ext_vector_type(2))) float v2f;
typedef __attribute__((ext_vector_type(8))) float v8f;

#define N_NODES 200000
#define FEATS   512
#define TOPK    5000
#define CAP     8192          // padded candidate buffer (power of 2 >= TOPK)

struct SelState {
    unsigned prefix;          // radix-select prefix (becomes exact threshold key)
    unsigned count_above;     // # keys strictly greater than current prefix region
    unsigned cand_cnt;        // compaction counter
    unsigned pad;
};

__device__ __forceinline__ unsigned flip_key(float f) {
    unsigned u = __float_as_uint(f);
    // larger unsigned key == larger float
    return (u & 0x80000000u) ? ~u : (u | 0x80000000u);
}

// ---------------------------------------------------------------------------
// Kernel 0: normalize scorer (scorer_n = scorer / ||scorer||), init state.
// ---------------------------------------------------------------------------
__global__ __launch_bounds__(256)
void prep_kernel(const float* __restrict__ scorer, float* __restrict__ scorer_n,
                 SelState* st) {
    __shared__ float red[256];
    int t = threadIdx.x;
    float v0 = scorer[t];
    float v1 = scorer[t + 256];
    red[t] = v0 * v0 + v1 * v1;
    __syncthreads();
    for (int s = 128; s > 0; s >>= 1) {
        if (t < s) red[t] += red[t + s];
        __syncthreads();
    }
    float inv = 1.0f / sqrtf(red[0]);
    scorer_n[t]       = v0 * inv;
    scorer_n[t + 256] = v1 * inv;
    if (t == 0) { st->prefix = 0u; st->count_above = 0u; st->cand_cnt = 0u; }
}

// ---------------------------------------------------------------------------
// Kernel 1: scores via V_WMMA_F32_16X16X4_F32. One wave -> 16 rows.
// A tile (16x4 f32): lanes 0-15 = rows, K={0,1}; lanes 16-31 = rows, K={2,3}.
// B tile (4x16 f32): scorer slice replicated across all 16 columns, so
// every column of D holds the score vector.
// ---------------------------------------------------------------------------
__global__ __launch_bounds__(128)
void score_kernel(const float* __restrict__ embs,
                  const float* __restrict__ scorer_n,
                  float* __restrict__ scores,
                  unsigned* __restrict__ keys) {
    int tile = blockIdx.x * 4 + (threadIdx.x >> 5);   // 0 .. 12499 (exact)
    int lane = threadIdx.x & 31;
    int R    = tile * 16;
    int row  = R + (lane & 15);
    int koff = (lane >> 4) * 2;                       // 0 for lanes 0-15, 2 for 16-31

    const float* arow = embs + (size_t)row * FEATS + koff;
    const float* brow = scorer_n + koff;

    v8f acc = {};
    #pragma unroll 4
    for (int k = 0; k < FEATS; k += 4) {
        v2f a = *(const v2f*)(arow + k);
        v2f b = *(const v2f*)(brow + k);
        acc = __builtin_amdgcn_wmma_f32_16x16x4_f32(
            false, a, false, b, (short)0, acc, false, false);
    }

    // lane t (t<8)    holds D[t][t]     in acc[t]  -> row R+t
    // lane 16+t (t<8) holds D[8+t][t]   in acc[t]  -> row R+8+t
    int t = lane & 7;
    float val = acc[0];
    val = (t == 1) ? acc[1] : val;
    val = (t == 2) ? acc[2] : val;
    val = (t == 3) ? acc[3] : val;
    val = (t == 4) ? acc[4] : val;
    val = (t == 5) ? acc[5] : val;
    val = (t == 6) ? acc[6] : val;
    val = (t == 7) ? acc[7] : val;

    bool doWrite = (lane < 8) || (lane >= 16 && lane < 24);
    int  m = (lane < 16) ? t : (t + 8);
    if (doWrite) {
        scores[R + m] = val;
        keys[R + m]   = flip_key(val);
    }
}

// ---------------------------------------------------------------------------
// Radix select: histogram / bin-pick per 8-bit digit, MSB first.
// ---------------------------------------------------------------------------
__global__ __launch_bounds__(256)
void zero_hist_kernel(unsigned* hist) { hist[threadIdx.x] = 0u; }

__global__ __launch_bounds__(256)
void hist_kernel(const unsigned* __restrict__ keys, int n,
                 const SelState* __restrict__ st, unsigned* __restrict__ hist,
                 int shift, unsigned himask) {
    unsigned pref = st->prefix;
    int i = blockIdx.x * blockDim.x + threadIdx.x;
    int stride = gridDim.x * blockDim.x;
    for (; i < n; i += stride) {
        unsigned k = keys[i];
        if (((k ^ pref) & himask) == 0u)
            atomicAdd(&hist[(k >> shift) & 0xFFu], 1u);
    }
}

__global__ void select_bin_kernel(const unsigned* __restrict__ hist,
                                  SelState* st, int shift, int K) {
    unsigned r = (unsigned)K - st->count_above;   // >= 1 by construction
    unsigned cum = 0;
    for (int b = 255; b >= 0; --b) {
        unsigned h = hist[b];
        if (cum + h >= r) {
            st->prefix |= ((unsigned)b) << shift;
            st->count_above += cum;
            return;
        }
        cum += h;
    }
}

// ---------------------------------------------------------------------------
// Compaction: all keys >= threshold -> composite (~key, index) for sorting.
// Ascending uint64 order == descending float, ties broken by smaller index.
// ---------------------------------------------------------------------------
__global__ __launch_bounds__(256)
void compact_kernel(const unsigned* __restrict__ keys, int n,
                    SelState* st, unsigned long long* __restrict__ cand) {
    unsigned T = st->prefix;
    int i = blockIdx.x * blockDim.x + threadIdx.x;
    int stride = gridDim.x * blockDim.x;
    for (; i < n; i += stride) {
        unsigned k = keys[i];
        if (k >= T) {
            unsigned pos = atomicAdd(&st->cand_cnt, 1u);
            if (pos < CAP)
                cand[pos] = ((unsigned long long)(~k) << 32) | (unsigned)i;
        }
    }
}

// ---------------------------------------------------------------------------
// Bitonic sort of CAP=8192 candidates in 64 KB LDS, then emit the first K
// (sorted descending by score, ascending by index) + tanh scale factors.
// ---------------------------------------------------------------------------
__global__ __launch_bounds__(1024)
void sort_kernel(const unsigned long long* __restrict__ cand,
                 const SelState* __restrict__ st,
                 const float* __restrict__ scores,
                 unsigned* __restrict__ sel_idx,
                 float* __restrict__ sel_scale) {
    __shared__ unsigned long long arr[CAP];   // 64 KB
    int n = (int)st->cand_cnt;
    if (n > CAP) n = CAP;
    for (int i = threadIdx.x; i < CAP; i += 1024)
        arr[i] = (i < n) ? cand[i] : ~0ull;
    __syncthreads();

    for (int k = 2; k <= CAP; k <<= 1) {
        for (int j = k >> 1; j > 0; j >>= 1) {
            for (int i = threadIdx.x; i < CAP; i += 1024) {
                int ixj = i ^ j;
                if (ixj > i) {
                    unsigned long long a = arr[i], b = arr[ixj];
                    bool up = ((i & k) == 0);
                    if ((a > b) == up) { arr[i] = b; arr[ixj] = a; }
                }
            }
            __syncthreads();
        }
    }

    for (int j = threadIdx.x; j < TOPK; j += 1024) {
        unsigned idx = (unsigned)(arr[j] & 0xFFFFFFFFu);
        sel_idx[j]   = idx;
        sel_scale[j] = tanhf(scores[idx]);
    }
}

// ---------------------------------------------------------------------------
// Output: out[f * K + j] = embs[idx_j][f] * tanh(score_j).  Coalesced stores;
// scattered row reads stay hot in the 192 MB L2 (gathered set is only 10 MB).
// ---------------------------------------------------------------------------
__global__ __launch_bounds__(256)
void write_out_kernel(const float* __restrict__ embs,
                      const unsigned* __restrict__ sel_idx,
                      const float* __restrict__ sel_scale,
                      float* __restrict__ out, int total) {
    int tid = blockIdx.x * 256 + threadIdx.x;
    if (tid >= total) return;
    int j = tid % TOPK;
    int f = tid / TOPK;
    unsigned idx = sel_idx[j];
    out[tid] = embs[(size_t)idx * FEATS + f] * sel_scale[j];
}

// ---------------------------------------------------------------------------
extern "C" void kernel_launch(void* const* d_in, const int* in_sizes, int n_in,
                              void* d_out, int out_size, void* d_ws, size_t ws_size,
                              hipStream_t stream) {
    const float* embs   = (const float*)d_in[0];   // [200000, 512]
    const float* scorer = (const float*)d_in[1];   // [512, 1]
    float* out = (float*)d_out;                    // [512, 5000]

    // workspace layout (bytes)
    char* ws = (char*)d_ws;
    unsigned*  keys      = (unsigned*)(ws);                       // 800000 B
    float*     scores    = (float*)(ws + 800000);                 // 800000 B
    unsigned long long* cand = (unsigned long long*)(ws + 1600000); // 65536 B (8-aligned)
    unsigned*  hist      = (unsigned*)(ws + 1665536);             // 1024 B
    SelState*  st        = (SelState*)(ws + 1666560);             // 16 B
    float*     scorer_n  = (float*)(ws + 1666576);                // 2048 B
    unsigned*  sel_idx   = (unsigned*)(ws + 1668624);             // 20000 B
    float*     sel_scale = (float*)(ws + 1688624);                // 20000 B

    // 0) normalize scorer + init state
    prep_kernel<<<1, 256, 0, stream>>>(scorer, scorer_n, st);

    // 1) scores via f32 WMMA: 12500 row-tiles, 4 waves/block -> 3125 blocks
    score_kernel<<<3125, 128, 0, stream>>>(embs, scorer_n, scores, keys);

    // 2) radix select the exact K-th threshold key (MSB-first, 4 passes)
    const int      shifts[4]  = {24, 16, 8, 0};
    const unsigned himasks[4] = {0u, 0xFF000000u, 0xFFFF0000u, 0xFFFFFF00u};
    for (int p = 0; p < 4; ++p) {
        zero_hist_kernel<<<1, 256, 0, stream>>>(hist);
        hist_kernel<<<512, 256, 0, stream>>>(keys, N_NODES, st, hist,
                                             shifts[p], himasks[p]);
        select_bin_kernel<<<1, 1, 0, stream>>>(hist, st, shifts[p], TOPK);
    }

    // 3) compact all keys >= threshold
    compact_kernel<<<512, 256, 0, stream>>>(keys, N_NODES, st, cand);

    // 4) single-workgroup bitonic sort in LDS, emit sorted top-K + scales
    sort_kernel<<<1, 1024, 0, stream>>>(cand, st, scores, sel_idx, sel_scale);

    // 5) gather + scale + transpose: 512*5000 = 2,560,000 elements
    write_out_kernel<<<10000, 256, 0, stream>>>(embs, sel_idx, sel_scale,
                                                out, FEATS * TOPK);
}